// Model_30975304139135
// MI455X (gfx1250) — compile-verified
//
#include <hip/hip_runtime.h>
#include <stdint.h>

// Problem constants (match reference)
#define Bc 4
#define Sc 512
#define Dc 768
#define Ec 12
#define Mc 4
#define Pc 132   // E*(E-1)

typedef unsigned int u32x4 __attribute__((ext_vector_type(4)));
typedef unsigned int u32x8 __attribute__((ext_vector_type(8)));

// p -> (head, tail) exactly like the reference pair enumeration:
// for h in range(E): for t in range(E) if t != h
__device__ __forceinline__ void pair_ht(int p, int& h, int& t) {
    h = p / (Ec - 1);
    int r = p - h * (Ec - 1);
    t = (r < h) ? r : r + 1;
}

// ---------------------------------------------------------------------------
// Kernel 1: per-(b,e) gather + small reductions
//   ent_att[b,e,s] = mean_m attention[b, mpos+1, s]
//   ent_emb[b,e,d] = logsumexp_m doc[b, mpos+1, d]
// ---------------------------------------------------------------------------
__global__ void prep_kernel(const float* __restrict__ doc,
                            const float* __restrict__ att,
                            const int*   __restrict__ mpos,
                            float* __restrict__ ea,
                            float* __restrict__ ee) {
    const int be = blockIdx.x;            // 0..B*E-1
    const int b  = be / Ec;

    const int i0 = mpos[be * Mc + 0] + 1;
    const int i1 = mpos[be * Mc + 1] + 1;
    const int i2 = mpos[be * Mc + 2] + 1;
    const int i3 = mpos[be * Mc + 3] + 1;

    const float* a = att + (size_t)b * Sc * Sc;
    for (int s = threadIdx.x; s < Sc; s += blockDim.x) {
        float v = a[(size_t)i0 * Sc + s] + a[(size_t)i1 * Sc + s] +
                  a[(size_t)i2 * Sc + s] + a[(size_t)i3 * Sc + s];
        ea[(size_t)be * Sc + s] = 0.25f * v;
    }

    const float* dp = doc + (size_t)b * Sc * Dc;
    for (int d = threadIdx.x; d < Dc; d += blockDim.x) {
        float v0 = dp[(size_t)i0 * Dc + d];
        float v1 = dp[(size_t)i1 * Dc + d];
        float v2 = dp[(size_t)i2 * Dc + d];
        float v3 = dp[(size_t)i3 * Dc + d];
        float m  = fmaxf(fmaxf(v0, v1), fmaxf(v2, v3));
        float s  = __expf(v0 - m) + __expf(v1 - m) + __expf(v2 - m) + __expf(v3 - m);
        ee[(size_t)be * Dc + d] = m + __logf(s);
    }
}

// ---------------------------------------------------------------------------
// Kernel 2: ctx[b,p,d] = logsumexp_s( w[b,p,s] * doc[b,s,d] )
// One block: (b, d-tile of 64, group of 12 pairs).
//   - doc tile (512x64 f32, 128 KB) staged by the Tensor Data Mover:
//     a single tensor_load_to_lds (TENSORcnt), issued by wave 0.
//   - ent_att[b] (12x512, 24 KB) staged via GLOBAL_LOAD_ASYNC_TO_LDS_B128
//     (ASYNCcnt).
// Tile reused 12 pairs x 2 passes = 24x. Writes output row 3 directly.
// ---------------------------------------------------------------------------
__global__ __launch_bounds__(256)
void ctx_kernel(const float* __restrict__ doc,
                const float* __restrict__ ea,
                float* __restrict__ out) {
    __shared__ float tile[Sc * 64];   // [s][dcol], 128 KB (TDM destination)
    __shared__ float eab[Ec * Sc];    // ent_att[b] slice, 24 KB
    __shared__ float wbuf[Sc];        // w[s] for current pair
    __shared__ float red[4][64];      // per-chunk partials
    __shared__ float colmax[64];      // combined max per column

    const int tid = threadIdx.x;
    const int dt  = blockIdx.x;       // 0..11 (d-tile)
    const int pg  = blockIdx.y;       // 0..10 (pair group of 12)
    const int b   = blockIdx.z;       // 0..3
    const int d0  = dt * 64;

    const float* src = doc + (size_t)b * Sc * Dc + d0;

    // ---- TDM: DMA doc[b, 0..511, d0..d0+63] -> LDS tile (one instruction) ----
    if (tid < 32) {
        const unsigned lds_base = (unsigned)(uintptr_t)tile;  // LDS byte offset
        const unsigned long long ga = (unsigned long long)(uintptr_t)src;
        u32x4 g0;
        g0[0] = 1u;                                            // count=1 valid
        g0[1] = lds_base;                                      // lds_addr
        g0[2] = (unsigned)ga;                                  // global_addr[31:0]
        g0[3] = (unsigned)((ga >> 32) & 0x1FFFFFFu) | 0x80000000u; // [56:32] | type=2
        u32x8 g1;
        g1[0] = 2u << 16;                                      // data_size = 4B
        g1[1] = ((unsigned)Dc & 0xFFFFu) << 16;                // tensor_dim0[15:0]
        g1[2] = ((unsigned)Dc >> 16) | (((unsigned)Sc & 0xFFFFu) << 16); // td0 hi | td1 lo
        g1[3] = ((unsigned)Sc >> 16) | (64u << 16);            // td1 hi | tile_dim0=64
        g1[4] = (unsigned)Sc;                                  // tile_dim1=512, tile_dim2=0
        g1[5] = (unsigned)Dc;                                  // tensor_dim0_stride[31:0]
        g1[6] = 0u;                                            // stride hi | dim1_stride lo
        g1[7] = 0u;
        asm volatile("tensor_load_to_lds %0, %1" :: "s"(g0), "s"(g1) : "memory");
    }

    // ---- async stage ent_att[b] (6144 floats) into LDS (b128 per lane) ----
    {
        const float* eb = ea + (size_t)b * Ec * Sc;
        const unsigned ebase = (unsigned)(uintptr_t)eab;
#pragma unroll
        for (int i = 0; i < 6; ++i) {
            int linear = i * 256 + tid;               // 0..1535 (x16B)
            const float* g = eb + (size_t)linear * 4;
            unsigned lds   = ebase + (unsigned)linear * 16u;
            asm volatile("global_load_async_to_lds_b128 %0, %1, off"
                         :: "v"(lds), "v"(g) : "memory");
        }
    }
    asm volatile("s_wait_asynccnt 0x0" ::: "memory");
    if (tid < 32) __builtin_amdgcn_s_wait_tensorcnt(0);
    __syncthreads();

    // wave32 mapping: wave&1 -> column half (conflict-free banks),
    // wave>>1 -> one of 4 s-chunks of 128
    const int wave  = tid >> 5;
    const int lane  = tid & 31;
    const int dcol  = (wave & 1) * 32 + lane;     // 0..63
    const int chunk = wave >> 1;                  // 0..3
    const int s0    = chunk * 128;

    for (int j = 0; j < 12; ++j) {
        const int p = pg * 12 + j;
        int h, t;
        pair_ht(p, h, t);
        for (int s = tid; s < Sc; s += 256)
            wbuf[s] = eab[h * Sc + s] * eab[t * Sc + s];
        __syncthreads();

        // pass 1: max over this chunk's 128 s values
        float m = -3.0e38f;
#pragma unroll 8
        for (int i = 0; i < 128; ++i) {
            float v = wbuf[s0 + i] * tile[(s0 + i) * 64 + dcol];
            m = fmaxf(m, v);
        }
        red[chunk][dcol] = m;
        __syncthreads();
        if (tid < 64)
            colmax[tid] = fmaxf(fmaxf(red[0][tid], red[1][tid]),
                                fmaxf(red[2][tid], red[3][tid]));
        __syncthreads();

        // pass 2: sum of exp(v - max)
        const float mm = colmax[dcol];
        float acc = 0.0f;
#pragma unroll 8
        for (int i = 0; i < 128; ++i) {
            float v = wbuf[s0 + i] * tile[(s0 + i) * 64 + dcol];
            acc += __expf(v - mm);
        }
        red[chunk][dcol] = acc;
        __syncthreads();
        if (tid < 64) {
            float ssum = red[0][tid] + red[1][tid] + red[2][tid] + red[3][tid];
            out[(((size_t)b * Pc + p) * 4 + 3) * Dc + d0 + tid] =
                colmax[tid] + __logf(ssum);
        }
        __syncthreads();   // protect wbuf/red before next pair
    }
}

// ---------------------------------------------------------------------------
// Kernel 3: fill output rows 0..2 (cls token, ent_emb[head], ent_emb[tail])
// ---------------------------------------------------------------------------
__global__ void assemble_kernel(const float* __restrict__ doc,
                                const float* __restrict__ ee,
                                float* __restrict__ out) {
    int g = blockIdx.x * blockDim.x + threadIdx.x;
    const int total = Bc * Pc * Dc;
    if (g >= total) return;
    int d  = g % Dc;
    int bp = g / Dc;
    int p  = bp % Pc;
    int b  = bp / Pc;
    int h, t;
    pair_ht(p, h, t);
    size_t base = (size_t)bp * 4 * Dc;
    out[base + d]            = doc[(size_t)b * Sc * Dc + d];          // cls (s=0)
    out[base + Dc + d]       = ee[((size_t)b * Ec + h) * Dc + d];
    out[base + 2 * Dc + d]   = ee[((size_t)b * Ec + t) * Dc + d];
}

// ---------------------------------------------------------------------------
extern "C" void kernel_launch(void* const* d_in, const int* in_sizes, int n_in,
                              void* d_out, int out_size, void* d_ws, size_t ws_size,
                              hipStream_t stream) {
    const float* doc  = (const float*)d_in[0];   // (B,S,D) f32
    const float* att  = (const float*)d_in[1];   // (B,S,S) f32
    const int*   mpos = (const int*)d_in[2];     // (B,E,M) i32
    float* out = (float*)d_out;                  // (B*P, 4, D) f32

    float* ea = (float*)d_ws;                    // (B,E,S)
    float* ee = ea + (size_t)Bc * Ec * Sc;       // (B,E,D)

    prep_kernel<<<Bc * Ec, 256, 0, stream>>>(doc, att, mpos, ea, ee);
    ctx_kernel<<<dim3(Dc / 64, Pc / 12, Bc), 256, 0, stream>>>(doc, ea, out);
    assemble_kernel<<<(Bc * Pc * Dc + 255) / 256, 256, 0, stream>>>(doc, ee, out);
}